// multiTimeAttention_46952582479931
// MI455X (gfx1250) — compile-verified
//
#include <hip/hip_runtime.h>

// Problem constants (from the reference)
#define B_   8
#define LQ_  128
#define LK_  512
#define ET_  128
#define H_   2
#define D_   64
#define NH_  128
#define ETK_ 64

typedef __attribute__((ext_vector_type(2))) float v2f;
typedef __attribute__((ext_vector_type(8))) float v8f;

// D = A(16x4, f32) * B(4x16, f32) + C(16x16, f32)  -- CDNA5 V_WMMA_F32_16X16X4_F32
__device__ __forceinline__ v8f wmma4(v2f a, v2f b, v8f c) {
    return __builtin_amdgcn_wmma_f32_16x16x4_f32(
        /*neg_a=*/false, a, /*neg_b=*/false, b,
        /*c_mod=*/(short)0, c, /*reuse_a=*/false, /*reuse_b=*/false);
}

// ---------------------------------------------------------------------------
// Generic tiled GEMM + bias: C[M x N] = A[M x K] * W[K x N] + bias[N]
// Row-major everywhere. One wave per 16x16 output tile, 4 waves per block.
// M, N, K all multiples of 16; grid.x = (M/16)*(N/16)/4.
// ---------------------------------------------------------------------------
__global__ void gemm_bias_kernel(const float* __restrict__ A,
                                 const float* __restrict__ W,
                                 const float* __restrict__ bias,
                                 float* __restrict__ C,
                                 int M, int N, int K) {
    const int lane = threadIdx.x & 31;
    const int wave = threadIdx.x >> 5;
    const int l16  = lane & 15;
    const int half = lane >> 4;            // 0: lanes 0-15, 1: lanes 16-31

    const int tilesN = N >> 4;
    const int tile   = blockIdx.x * 4 + wave;
    const int tm     = (tile / tilesN) << 4;
    const int tn     = (tile % tilesN) << 4;

    v8f acc = {};
    // A lane view: row = tm + l16, K pair starting at 2*half
    const float* arow = A + (tm + l16) * K + 2 * half;
    // B lane view: col = tn + l16, rows (k0 + 2*half), (k0 + 2*half + 1)
    const float* bcol = W + 2 * half * N + tn + l16;

    for (int k0 = 0; k0 < K; k0 += 4) {
        v2f a = *(const v2f*)(arow + k0);
        v2f b;
        b.x = bcol[k0 * N];
        b.y = bcol[k0 * N + N];
        acc = wmma4(a, b, acc);
    }

    const float bv  = bias[tn + l16];
    const int   row = tm + 8 * half;
#pragma unroll
    for (int r = 0; r < 8; ++r)
        C[(row + r) * N + tn + l16] = acc[r] + bv;
}

// ---------------------------------------------------------------------------
// Scores + row-max + exp:  E[b,h,q,k] = exp( (Q_h . K_h)/8 - rowmax )
// One block per (b, h, 16-row strip of q). 256 threads = 8 waves; wave w owns
// k-range [w*64, w*64+64) as 4 WMMA N-tiles. Row max reduced via shfl + LDS.
// ---------------------------------------------------------------------------
__global__ void scores_exp_kernel(const float* __restrict__ Qp,
                                  const float* __restrict__ Kp,
                                  float* __restrict__ E) {
    const int lane = threadIdx.x & 31;
    const int wave = threadIdx.x >> 5;     // 0..7
    const int l16  = lane & 15;
    const int half = lane >> 4;

    const int bidx = blockIdx.x;
    const int mt   = bidx & 7;             // LQ/16 = 8 strips
    const int h    = (bidx >> 3) & 1;
    const int b    = bidx >> 4;

    // A: Q rows (tm+l16), features e in [h*64, h*64+64)
    const float* qrow = Qp + (b * LQ_ + mt * 16 + l16) * ET_ + h * ETK_ + 2 * half;

    v8f acc[4];
#pragma unroll
    for (int nt = 0; nt < 4; ++nt) acc[nt] = (v8f){0,0,0,0,0,0,0,0};

    for (int e0 = 0; e0 < ETK_; e0 += 4) {
        v2f a = *(const v2f*)(qrow + e0);
#pragma unroll
        for (int nt = 0; nt < 4; ++nt) {
            const int kcol = wave * 64 + nt * 16 + l16;   // B column = key index
            const float* kp = Kp + (b * LK_ + kcol) * ET_ + h * ETK_ + e0 + 2 * half;
            v2f bb = *(const v2f*)kp;                      // K^T: (e, e+1) pair
            acc[nt] = wmma4(a, bb, acc[nt]);
        }
    }

    // scale by 1/sqrt(64) and find per-row max across the full 512-key row
    float s[4][8];
    float lmax[8];
#pragma unroll
    for (int r = 0; r < 8; ++r) {
        float m = -3.4e38f;
#pragma unroll
        for (int nt = 0; nt < 4; ++nt) {
            float v = acc[nt][r] * 0.125f;
            s[nt][r] = v;
            m = fmaxf(m, v);
        }
        lmax[r] = m;
    }
    // reduce within each 16-lane group (same rows share a group)
#pragma unroll
    for (int r = 0; r < 8; ++r) {
        for (int m = 1; m < 16; m <<= 1)
            lmax[r] = fmaxf(lmax[r], __shfl_xor(lmax[r], m, 32));
    }

    __shared__ float pmax[8][16];
    if (l16 == 0) {
#pragma unroll
        for (int r = 0; r < 8; ++r)
            pmax[wave][half * 8 + r] = lmax[r];
    }
    __syncthreads();

    float rm[8];
#pragma unroll
    for (int r = 0; r < 8; ++r) {
        float m = pmax[0][half * 8 + r];
#pragma unroll
        for (int w = 1; w < 8; ++w)
            m = fmaxf(m, pmax[w][half * 8 + r]);
        rm[r] = m;
    }

    float* erow = E + ((b * H_ + h) * LQ_ + mt * 16 + 8 * half) * LK_ + wave * 64 + l16;
#pragma unroll
    for (int r = 0; r < 8; ++r) {
#pragma unroll
        for (int nt = 0; nt < 4; ++nt)
            erow[r * LK_ + nt * 16] = __expf(s[nt][r] - rm[r]);
    }
}

// ---------------------------------------------------------------------------
// Attention contraction: for each (b,h):
//   num = E[128x512] @ (mask*value)[512x64]
//   den = E[128x512] @ mask[512x64]
//   X[b, q, h*64+d] = num/den
// One block per (b, h, 16-row strip); 4 waves, wave w owns d-tile [w*16, w*16+16).
// ---------------------------------------------------------------------------
__global__ void attn_kernel(const float* __restrict__ E,
                            const float* __restrict__ value,
                            const int* __restrict__ mask,
                            float* __restrict__ X) {
    const int lane = threadIdx.x & 31;
    const int wave = threadIdx.x >> 5;     // 0..3 -> d tile
    const int l16  = lane & 15;
    const int half = lane >> 4;

    const int bidx = blockIdx.x;
    const int mt   = bidx & 7;
    const int h    = (bidx >> 3) & 1;
    const int b    = bidx >> 4;

    const int d = wave * 16 + l16;         // B column = feature d
    const float* erow = E + ((b * H_ + h) * LQ_ + mt * 16 + l16) * LK_ + 2 * half;

    v8f accN = (v8f){0,0,0,0,0,0,0,0};
    v8f accD = (v8f){0,0,0,0,0,0,0,0};

    for (int k0 = 0; k0 < LK_; k0 += 4) {
        v2f a = *(const v2f*)(erow + k0);
        const int kk = k0 + 2 * half;
        const int i0 = (b * LK_ + kk) * D_ + d;
        const float m0 = (float)mask[i0];
        const float m1 = (float)mask[i0 + D_];
        v2f bD, bN;
        bD.x = m0;               bD.y = m1;
        bN.x = m0 * value[i0];   bN.y = m1 * value[i0 + D_];
        accN = wmma4(a, bN, accN);
        accD = wmma4(a, bD, accD);
    }

    const int row = mt * 16 + 8 * half;
#pragma unroll
    for (int r = 0; r < 8; ++r)
        X[(b * LQ_ + row + r) * NH_ + h * D_ + d] = accN[r] / accD[r];
}

// ---------------------------------------------------------------------------
extern "C" void kernel_launch(void* const* d_in, const int* in_sizes, int n_in,
                              void* d_out, int out_size, void* d_ws, size_t ws_size,
                              hipStream_t stream) {
    (void)in_sizes; (void)n_in; (void)out_size; (void)ws_size;

    const float* query = (const float*)d_in[0];
    const float* key   = (const float*)d_in[1];
    const float* value = (const float*)d_in[2];
    const int*   mask  = (const int*)  d_in[3];
    const float* Wq    = (const float*)d_in[4];
    const float* bq    = (const float*)d_in[5];
    const float* Wk    = (const float*)d_in[6];
    const float* bk    = (const float*)d_in[7];
    const float* Wo    = (const float*)d_in[8];
    const float* bo    = (const float*)d_in[9];
    float* out = (float*)d_out;

    // workspace layout (floats): Qp | Kp | E | X  == 7 MB total
    float* ws = (float*)d_ws;
    float* Qp = ws;                                  // 1024*128
    float* Kp = Qp + (B_ * LQ_ * ET_);               // 4096*128
    float* E  = Kp + (B_ * LK_ * ET_);               // 8*2*128*512
    float* X  = E  + (B_ * H_ * LQ_ * LK_);          // 1024*128

    // Q/K projections: tiles = (M/16)*(N/16), 4 waves (tiles) per 128-thread block
    gemm_bias_kernel<<<(B_*LQ_/16)*(ET_/16)/4, 128, 0, stream>>>(query, Wq, bq, Qp,
                                                                 B_*LQ_, ET_, ET_);
    gemm_bias_kernel<<<(B_*LK_/16)*(ET_/16)/4, 128, 0, stream>>>(key,   Wk, bk, Kp,
                                                                 B_*LK_, ET_, ET_);
    // scores + exp: one block per (b, h, 16-q strip)
    scores_exp_kernel<<<B_*H_*(LQ_/16), 256, 0, stream>>>(Qp, Kp, E);
    // masked attention contraction (num/den GEMMs + divide)
    attn_kernel<<<B_*H_*(LQ_/16), 128, 0, stream>>>(E, value, mask, X);
    // output projection
    gemm_bias_kernel<<<(B_*LQ_/16)*(NH_/16)/4, 128, 0, stream>>>(X, Wo, bo, out,
                                                                 B_*LQ_, NH_, NH_);
}